// SimpleColorDecoder_68238440399516
// MI455X (gfx1250) — compile-verified
//
#include <hip/hip_runtime.h>

typedef __attribute__((ext_vector_type(2))) float v2f;
typedef __attribute__((ext_vector_type(8))) float v8f;

#define BATCH 8
#define CIN   256
#define KCLS  21
#define SMALL 4096     // 64*64
#define BIG   262144   // 512*512

// ---------------------------------------------------------------------------
// Kernel 1: logits[b][k][s] = sum_c W[k][c] * fm[b][c][s] + bias[k]
// One wave (32 lanes) per 16-pixel tile; two 16x16 M-tiles (classes 0..31,
// rows >= 21 zero padded) accumulated with V_WMMA_F32_16X16X4_F32, sharing
// the B fragment. Zero padding of the second A tile is branchless (clamped
// row + 0/1 mask multiply) so the hot loop has no exec-mask branches.
//
// f32 WMMA fragment layouts (ISA 7.12.2):
//   A 16x4:  lane<16 -> row M=lane, {v0,v1} = K {0,1}; lane>=16 -> K {2,3}
//   B 4x16:  lane<16 -> col N=lane, {v0,v1} = K {0,1}; lane>=16 -> K {2,3}
//   C/D 16x16: vgpr v, lane<16 -> (M=v,   N=lane)
//                      lane>=16 -> (M=v+8, N=lane-16)
// ---------------------------------------------------------------------------
__global__ __launch_bounds__(32)
void k_gemm_logits(const float* __restrict__ fm,
                   const float* __restrict__ Wc,
                   const float* __restrict__ bc,
                   float* __restrict__ logits) {
  const int tile = blockIdx.x;        // 0..255  (16-pixel tile)
  const int b    = blockIdx.y;        // 0..7
  const int lane = threadIdx.x;       // 0..31
  const int half = lane >> 4;         // 0/1
  const int l15  = lane & 15;
  const int pixbase = tile * 16;
  const float* fmb = fm + (size_t)b * CIN * SMALL;

  // Loop-invariant A-row bases (branchless zero pad for rows >= 21).
  const float* arow0 = Wc + l15 * CIN;
  const int    m1    = 16 + l15;
  const float  mask1 = (m1 < KCLS) ? 1.0f : 0.0f;
  const float* arow1 = Wc + min(m1, KCLS - 1) * CIN;

  v8f acc0 = {};   // classes 0..15
  v8f acc1 = {};   // classes 16..31 (>=21 are zero rows)

  for (int k0 = 0; k0 < CIN; k0 += 4) {
    const int kk = k0 + 2 * half;

    v2f bfrag;
    bfrag.x = fmb[(size_t)kk       * SMALL + pixbase + l15];
    bfrag.y = fmb[(size_t)(kk + 1) * SMALL + pixbase + l15];

    v2f a0;                       // adjacent -> single b64 load
    a0.x = arow0[kk];
    a0.y = arow0[kk + 1];

    v2f a1;                       // unconditional b64 load, masked to zero
    a1.x = mask1 * arow1[kk];
    a1.y = mask1 * arow1[kk + 1];

    acc0 = __builtin_amdgcn_wmma_f32_16x16x4_f32(false, a0, false, bfrag,
                                                 (short)0, acc0, false, false);
    acc1 = __builtin_amdgcn_wmma_f32_16x16x4_f32(false, a1, false, bfrag,
                                                 (short)0, acc1, false, false);
  }

  const int pix = pixbase + l15;
  float* lb = logits + (size_t)b * KCLS * SMALL;
  for (int v = 0; v < 8; ++v) {
    const int c0 = v + 8 * half;                     // always < 16
    lb[(size_t)c0 * SMALL + pix] = acc0[v] + bc[c0];
    const int c1 = 16 + v + 8 * half;
    if (c1 < KCLS)
      lb[(size_t)c1 * SMALL + pix] = acc1[v] + bc[c1];
  }
}

// ---------------------------------------------------------------------------
// Kernel 2: softmax over the 21 classes, in place. One thread per (b, pixel).
// ---------------------------------------------------------------------------
__global__ __launch_bounds__(256)
void k_softmax(float* __restrict__ seg) {
  const int idx = blockIdx.x * blockDim.x + threadIdx.x;   // b*4096 + s
  if (idx >= BATCH * SMALL) return;
  const int b = idx >> 12;
  const int s = idx & (SMALL - 1);
  float* p = seg + (size_t)b * KCLS * SMALL + s;

  float vals[KCLS];
  float m = -1e30f;
  for (int k = 0; k < KCLS; ++k) {
    vals[k] = p[(size_t)k * SMALL];
    m = fmaxf(m, vals[k]);
  }
  float sum = 0.0f;
  for (int k = 0; k < KCLS; ++k) {
    vals[k] = __expf(vals[k] - m);
    sum += vals[k];
  }
  const float inv = 1.0f / sum;
  for (int k = 0; k < KCLS; ++k)
    p[(size_t)k * SMALL] = vals[k] * inv;
}

// ---------------------------------------------------------------------------
// Kernel 3: adjoint of align_corners bilinear upsample applied to x:
//   u[b][c][sy][sx] = sum_P w(P, s) * x[b][c][P]
// so that sum_P x[P]*seg_up[P] == sum_s u[s]*seg[s]  (bilinear is linear).
// One thread per (b, c, small-cell); ~17x17 separable gather.
// ---------------------------------------------------------------------------
__global__ __launch_bounds__(256)
void k_adjoint(const float* __restrict__ x, float* __restrict__ u) {
  const int idx = blockIdx.x * blockDim.x + threadIdx.x;   // over 8*3*4096
  if (idx >= BATCH * 3 * SMALL) return;
  const int s   = idx & (SMALL - 1);
  const int bc  = idx >> 12;            // b*3 + c
  const int sy  = s >> 6;
  const int sx  = s & 63;
  const float* xp = x + (size_t)bc * BIG;

  const float scale = 511.0f / 63.0f;   // big samples per small cell
  const float inv   = 63.0f / 511.0f;   // small coord per big pixel

  int yLo = max(0,   (int)ceilf((float)(sy - 1) * scale));
  int yHi = min(511, (int)ceilf((float)(sy + 1) * scale) - 1);
  int xLo = max(0,   (int)ceilf((float)(sx - 1) * scale));
  int xHi = min(511, (int)ceilf((float)(sx + 1) * scale) - 1);
  const int ny = yHi - yLo + 1;
  const int nx = xHi - xLo + 1;

  float wy[20], wx[20];
  for (int i = 0; i < ny; ++i) {
    const float yf = (float)(yLo + i) * inv;
    const float y0 = floorf(yf);
    const float f  = yf - y0;
    const int  iy0 = (int)y0;
    wy[i] = (iy0 == sy) ? (1.0f - f) : ((iy0 == sy - 1) ? f : 0.0f);
  }
  for (int j = 0; j < nx; ++j) {
    const float xf = (float)(xLo + j) * inv;
    const float x0 = floorf(xf);
    const float f  = xf - x0;
    const int  ix0 = (int)x0;
    wx[j] = (ix0 == sx) ? (1.0f - f) : ((ix0 == sx - 1) ? f : 0.0f);
  }

  float acc = 0.0f;
  for (int i = 0; i < ny; ++i) {
    const float wyi = wy[i];
    if (wyi == 0.0f) continue;
    const float* row = xp + (size_t)(yLo + i) * 512 + xLo;
    float r = 0.0f;
    for (int j = 0; j < nx; ++j) r += wx[j] * row[j];
    acc += wyi * r;
  }
  u[idx] = acc;
}

// ---------------------------------------------------------------------------
// Kernel 4: q[b][k][c] = (1/BIG) * sum_s seg[b][k][s] * u[b][c][s]
// One 256-thread block per (b, k); LDS tree reduction of 3 channels.
// ---------------------------------------------------------------------------
__global__ __launch_bounds__(256)
void k_q(const float* __restrict__ seg, const float* __restrict__ u,
         float* __restrict__ q) {
  const int bk = blockIdx.x;            // 0..167
  const int b  = bk / KCLS;
  const float* sp = seg + (size_t)bk * SMALL;
  const float* up = u + (size_t)b * 3 * SMALL;

  float a0 = 0.0f, a1 = 0.0f, a2 = 0.0f;
  for (int s = threadIdx.x; s < SMALL; s += 256) {
    const float sg = sp[s];
    a0 += sg * up[s];
    a1 += sg * up[SMALL + s];
    a2 += sg * up[2 * SMALL + s];
  }

  __shared__ float r0[256], r1[256], r2[256];
  r0[threadIdx.x] = a0; r1[threadIdx.x] = a1; r2[threadIdx.x] = a2;
  __syncthreads();
  for (int off = 128; off > 0; off >>= 1) {
    if (threadIdx.x < off) {
      r0[threadIdx.x] += r0[threadIdx.x + off];
      r1[threadIdx.x] += r1[threadIdx.x + off];
      r2[threadIdx.x] += r2[threadIdx.x + off];
    }
    __syncthreads();
  }
  if (threadIdx.x == 0) {
    const float norm = 1.0f / (float)BIG;
    q[bk * 3 + 0] = r0[0] * norm;
    q[bk * 3 + 1] = r1[0] * norm;
    q[bk * 3 + 2] = r2[0] * norm;
  }
}

// ---------------------------------------------------------------------------
// Kernel 5: out[b][k][P] = sum_c x[b][c][P] * q[b][k][c]
// Pure-bandwidth kernel: 176 MB coalesced nontemporal stores (output is
// written once and never re-read on-device -> NT hint keeps L2 clean),
// q staged in LDS.
// ---------------------------------------------------------------------------
__global__ __launch_bounds__(256)
void k_out(const float* __restrict__ x, const float* __restrict__ q,
           float* __restrict__ out) {
  __shared__ float qs[KCLS * 3];
  const int b = blockIdx.y;
  const int P = blockIdx.x * 256 + threadIdx.x;
  if (threadIdx.x < KCLS * 3) qs[threadIdx.x] = q[b * KCLS * 3 + threadIdx.x];
  __syncthreads();

  const float* xb = x + (size_t)b * 3 * BIG;
  const float x0 = __builtin_nontemporal_load(&xb[P]);
  const float x1 = __builtin_nontemporal_load(&xb[BIG + P]);
  const float x2 = __builtin_nontemporal_load(&xb[2 * BIG + P]);

  float* ob = out + (size_t)b * KCLS * BIG + P;
  for (int k = 0; k < KCLS; ++k) {
    const float v = x0 * qs[k * 3 + 0] + x1 * qs[k * 3 + 1]
                  + x2 * qs[k * 3 + 2];
    __builtin_nontemporal_store(v, &ob[(size_t)k * BIG]);
  }
}

// ---------------------------------------------------------------------------
extern "C" void kernel_launch(void* const* d_in, const int* in_sizes, int n_in,
                              void* d_out, int out_size, void* d_ws, size_t ws_size,
                              hipStream_t stream) {
  (void)in_sizes; (void)n_in; (void)out_size; (void)ws_size;
  const float* fm = (const float*)d_in[0];   // [8,256,64,64]
  const float* x  = (const float*)d_in[1];   // [8,3,512,512]
  const float* Wc = (const float*)d_in[2];   // [21,256]
  const float* bc = (const float*)d_in[3];   // [21]
  float* out = (float*)d_out;                // [8,21,512,512]

  float* seg = (float*)d_ws;                       // 8*21*4096 floats (logits -> seg)
  float* u   = seg + (size_t)BATCH * KCLS * SMALL; // 8*3*4096 floats
  float* q   = u   + (size_t)BATCH * 3 * SMALL;    // 8*21*3 floats

  dim3 g1(SMALL / 16, BATCH);
  k_gemm_logits<<<g1, 32, 0, stream>>>(fm, Wc, bc, seg);

  k_softmax<<<(BATCH * SMALL + 255) / 256, 256, 0, stream>>>(seg);

  k_adjoint<<<(BATCH * 3 * SMALL + 255) / 256, 256, 0, stream>>>(x, u);

  k_q<<<BATCH * KCLS, 256, 0, stream>>>(seg, u, q);

  dim3 g5(BIG / 256, BATCH);
  k_out<<<g5, 256, 0, stream>>>(x, q, out);
}